// GeometricSuperpositionSearch_10565619548964
// MI455X (gfx1250) — compile-verified
//
#include <hip/hip_runtime.h>
#include <hip/hip_bf16.h>
#include <stdint.h>

// ---------------------------------------------------------------------------
// GeometricSuperpositionSearch on MI455X (gfx1250, wave32, WMMA)
//
// Key identity: gp(M, x) and gp(y, Mr) are linear in x/y, so the sandwich
// outcomes = gp(gp(M, x), rev(M)) == x @ T  with a per-(b,k) 16x16 matrix T.
// The hot loop becomes (8192x16)@(16x16) fp32 GEMMs -> V_WMMA_F32_16X16X4_F32.
// ---------------------------------------------------------------------------

typedef float v2f __attribute__((ext_vector_type(2)));
typedef float v8f __attribute__((ext_vector_type(8)));

#define GSS_B   32
#define GSS_N   8192
#define GSS_K   8
#define GSS_R   8
#define GSS_D   16
#define GSS_CTRL 8

// basis-blade ordering: masks sorted by (popcount, value)
__constant__ int   c_IDX[16]   = {0,1,2,5,3,6,7,11,4,8,9,12,10,13,14,15}; // mask -> index
__constant__ int   c_GRADE[16] = {0,1,1,1,1,2,2,2,2,2,2,3,3,3,3,4};       // index -> grade
__constant__ float c_REV[16]   = {1,1,1,1,1,-1,-1,-1,-1,-1,-1,-1,-1,-1,-1,1};

__device__ __forceinline__ int cayley_sign(int a, int bm) {
    int swaps = 0;
    int t = a >> 1;
    while (t) { swaps += __popc(t & bm); t >>= 1; }
    int sign = (swaps & 1) ? -1 : 1;
    if ((a & bm) & 1) return 0;   // SQ[0] == 0: e0^2 = 0 annihilates
    return sign;                  // SQ[1..3] == +1
}

__device__ __forceinline__ uint32_t hash3(uint32_t a, uint32_t b, uint32_t c) {
    uint32_t x = a * 0x9E3779B1u ^ b * 0x85EBCA77u ^ c * 0xC2B2AE3Du;
    x ^= x >> 16; x *= 0x7FEB352Du;
    x ^= x >> 15; x *= 0x846CA68Bu;
    x ^= x >> 16;
    return x;
}

// ---------------------------------------------------------------------------
// Kernel 1: per-batch means (cpu_summary over N, rule_summary over R)
// ---------------------------------------------------------------------------
__global__ __launch_bounds__(256)
void gss_reduce(const float* __restrict__ x,      // (B,N,16)
                const float* __restrict__ rmem,   // (B,R,16)
                float* __restrict__ cpu_sum,      // (B,16)
                float* __restrict__ rule_sum)     // (B,16)
{
    __shared__ float part[256];
    const int b = blockIdx.x;
    const int t = threadIdx.x;
    const int d = t & 15;          // column 0..15
    const int g = t >> 4;          // row-group 0..15
    const float* xb = x + (size_t)b * GSS_N * GSS_D;

    float acc = 0.f;
    for (int j = 0; j < GSS_N / 16; ++j) {
        acc += xb[(size_t)(g + 16 * j) * GSS_D + d];
    }
    part[t] = acc;
    __syncthreads();

    if (t < 16) {
        float s = 0.f;
        for (int gg = 0; gg < 16; ++gg) s += part[gg * 16 + t];
        cpu_sum[b * 16 + t] = s * (1.0f / GSS_N);

        float rs = 0.f;
        for (int r = 0; r < GSS_R; ++r)
            rs += rmem[(size_t)b * GSS_R * GSS_D + r * GSS_D + t];
        rule_sum[b * 16 + t] = rs * (1.0f / GSS_R);
    }
}

// ---------------------------------------------------------------------------
// Kernel 2: per-batch setup — grade norms, MLP biases, templates+rule mod,
//           Cayley table, and the 8 sandwich matrices T_bk (16x16 each).
// ---------------------------------------------------------------------------
__global__ __launch_bounds__(256)
void gss_prep(const float* __restrict__ ctrl,      // (B,8)
              const float* __restrict__ templates, // (K,16)
              const float* __restrict__ w1,        // (13,64)
              const float* __restrict__ b1,        // (64)
              const float* __restrict__ w2,        // (64,8)
              const float* __restrict__ b2,        // (8)
              const float* __restrict__ cell_w,    // (16,8)
              const float* __restrict__ cell_b,    // (8)
              const float* __restrict__ rsp_w,     // (16,8)
              const float* __restrict__ rsp_b,     // (8)
              const float* __restrict__ rp_w,      // (16,128)
              const float* __restrict__ rp_b,      // (128)
              const float* __restrict__ cpu_sum,   // (B,16)
              const float* __restrict__ rule_sum,  // (B,16)
              float* __restrict__ Tout,            // (B,8,16,16)
              float* __restrict__ wpad,            // (16,16)  cell_w zero-padded
              float* __restrict__ colbias)         // (B,16)
{
    __shared__ float feat[13];
    __shared__ float h[64];
    __shared__ float rsumS[16];
    __shared__ float tmplS[8][16];
    __shared__ float sgn[16][16];
    __shared__ int   kidx[16][16];
    __shared__ float A[16][16];
    __shared__ float Bm[16][16];

    const int b = blockIdx.x;
    const int t = threadIdx.x;

    if (t < 5) {
        float s = 1e-9f;
        for (int i = 0; i < 16; ++i) {
            if (c_GRADE[i] == t) { float v = cpu_sum[b * 16 + i]; s += v * v; }
        }
        feat[t] = sqrtf(s);
    }
    if (t >= 5 && t < 13) feat[t] = ctrl[b * GSS_CTRL + (t - 5)];
    if (t < 16) rsumS[t] = rule_sum[b * 16 + t];
    __syncthreads();

    if (t < 64) {
        float s = b1[t];
        for (int j = 0; j < 13; ++j) s += feat[j] * w1[j * 64 + t];
        h[t] = fmaxf(s, 0.f);
    }
    __syncthreads();

    // Cayley table (all 256 threads: one (a_mask, b_mask) pair each)
    {
        int am = t >> 4, bm = t & 15;
        kidx[c_IDX[am]][c_IDX[bm]] = c_IDX[am ^ bm];
        sgn [c_IDX[am]][c_IDX[bm]] = (float)cayley_sign(am, bm);
    }
    // templates + rule modulation
    if (t < 128) {
        int k = t >> 4, d = t & 15;
        float s = templates[k * 16 + d] + rp_b[k * 16 + d];
        for (int i = 0; i < 16; ++i) s += rsumS[i] * rp_w[i * 128 + k * 16 + d];
        tmplS[k][d] = s;
    }
    // per-(b,k) score column bias: cell_b + global_bias + rule-score proj
    if (t < 16) {
        float cb = 0.f;
        if (t < 8) {
            float gb = b2[t];
            for (int o = 0; o < 64; ++o) gb += h[o] * w2[o * 8 + t];
            float rp = rsp_b[t];
            for (int i = 0; i < 16; ++i) rp += rsumS[i] * rsp_w[i * 8 + t];
            cb = cell_b[t] + gb + rp;
        }
        colbias[b * 16 + t] = cb;
    }
    // zero-padded cell_w as a 16x16 WMMA B-operand (written once)
    if (b == 0) {
        int d = t >> 4, k = t & 15;
        wpad[d * 16 + k] = (k < 8) ? cell_w[d * 8 + k] : 0.f;
    }
    __syncthreads();

    // Build T_bk = A_bk @ B_bk  where  tmp = x@A (A[j][c] = sum_i M_i C[i][j][c])
    //                                 out = tmp@B (B[c][l] = sum_j Mr_j C[c][j][l])
    for (int k = 0; k < GSS_K; ++k) {
        int r = t >> 4, c = t & 15;
        float av = 0.f, bv = 0.f;
        for (int i = 0; i < 16; ++i) {
            if (kidx[i][r] == c) av += sgn[i][r] * tmplS[k][i];          // A[r][c]
            if (kidx[r][i] == c) bv += sgn[r][i] * tmplS[k][i] * c_REV[i]; // Bm[r][c]
        }
        A[r][c]  = av;
        Bm[r][c] = bv;
        __syncthreads();
        float tv = 0.f;
        for (int m = 0; m < 16; ++m) tv += A[r][m] * Bm[m][c];
        Tout[(((size_t)b * GSS_K + k) * 16 + r) * 16 + c] = tv;
        __syncthreads();
    }
}

// ---------------------------------------------------------------------------
// Kernel 3: main pass. One wave32 per 16-row tile.
//   logits  = X_tile(16x16) @ Wpad(16x16)   (4x chained V_WMMA_F32_16X16X4_F32)
//   weights = softmax(logits + bias + gumbel) over k (cross-lane, 8 cols)
//   out     = sum_k diag(w_k) * (X_tile @ T_bk)   (8 * 4 chained WMMAs)
// ---------------------------------------------------------------------------
#define TILES_PER_B   (GSS_N / 16)   // 512
#define WAVES_PER_BLK 4
#define BLKS_PER_B    (TILES_PER_B / WAVES_PER_BLK) // 128

__global__ __launch_bounds__(WAVES_PER_BLK * 32)
void gss_main(const float* __restrict__ x,        // (B,N,16)
              const float* __restrict__ T,        // (B,8,16,16)
              const float* __restrict__ wpad,     // (16,16)
              const float* __restrict__ colbias,  // (B,16)
              float* __restrict__ out)            // (B,N,16)
{
    __shared__ float sT[GSS_K * 256];
    __shared__ float sW[256];
    __shared__ float sCB[16];

    const int b  = blockIdx.x / BLKS_PER_B;
    const int tg = blockIdx.x % BLKS_PER_B;

    const float* Tb = T + (size_t)b * GSS_K * 256;
    for (int i = threadIdx.x; i < GSS_K * 256; i += blockDim.x) sT[i] = Tb[i];
    for (int i = threadIdx.x; i < 256; i += blockDim.x) sW[i] = wpad[i];
    if (threadIdx.x < 16) sCB[threadIdx.x] = colbias[b * 16 + threadIdx.x];
    __syncthreads();

    const int wave = threadIdx.x >> 5;
    const int lane = threadIdx.x & 31;
    const int tile = tg * WAVES_PER_BLK + wave;
    const int n0   = tile * 16;

    const float* xb = x   + ((size_t)b * GSS_N + n0) * GSS_D;
    float*       ob = out + ((size_t)b * GSS_N + n0) * GSS_D;

    const int row  = lane & 15;   // M-row for A-frag, also N-col for B/C/D frags
    const int half = lane >> 4;
    const int col  = row;

    // A-fragments: 16x4 f32 per chunk; lanes 0-15 hold K={4c,4c+1},
    // lanes 16-31 hold K={4c+2,4c+3}  (2 VGPRs / lane) — exact 1x read traffic.
    v2f a[4];
    #pragma unroll
    for (int c = 0; c < 4; ++c) {
        a[c] = *(const v2f*)(xb + row * 16 + c * 4 + 2 * half);
    }

    // ---- scores = X @ Wpad + colbias (per column) ----
    float cb = sCB[col];
    v8f sc;
    #pragma unroll
    for (int i = 0; i < 8; ++i) sc[i] = cb;
    #pragma unroll
    for (int c = 0; c < 4; ++c) {
        v2f bw;
        bw.x = sW[(c * 4 + 2 * half)     * 16 + col];
        bw.y = sW[(c * 4 + 2 * half + 1) * 16 + col];
        sc = __builtin_amdgcn_wmma_f32_16x16x4_f32(
                 false, a[c], false, bw, (short)0, sc, false, false);
    }

    // ---- gumbel + softmax over k (columns 0..7, cross-lane in 8-lane groups) ----
    v8f w;
    #pragma unroll
    for (int v = 0; v < 8; ++v) {
        const int rrow = v + 8 * half;
        float s;
        if (col < 8) {
            uint32_t hsh = hash3((uint32_t)b, (uint32_t)(n0 + rrow), (uint32_t)col);
            float u = fmaxf((float)(hsh >> 8) * (1.0f / 16777216.0f), 1e-9f);
            float g = -__logf(-__logf(u));
            s = sc[v] + g;           // tau = clip(1.0, 0.1, 5.0) = 1.0
        } else {
            s = -1e30f;              // padded columns never win / never read
        }
        float m = s;
        #pragma unroll
        for (int mk = 1; mk < 8; mk <<= 1) m = fmaxf(m, __shfl_xor(m, mk, 32));
        float e = __expf(s - m);
        float se = e;
        #pragma unroll
        for (int mk = 1; mk < 8; mk <<= 1) se += __shfl_xor(se, mk, 32);
        w[v] = e / se;
    }

    // ---- out = sum_k w_k ⊙ (X @ T_bk) ----
    v8f acc;
    #pragma unroll
    for (int i = 0; i < 8; ++i) acc[i] = 0.f;

    #pragma unroll
    for (int k = 0; k < GSS_K; ++k) {
        const float* Tk = sT + k * 256;
        v8f y;
        #pragma unroll
        for (int i = 0; i < 8; ++i) y[i] = 0.f;
        #pragma unroll
        for (int c = 0; c < 4; ++c) {
            v2f bt;
            bt.x = Tk[(c * 4 + 2 * half)     * 16 + col];
            bt.y = Tk[(c * 4 + 2 * half + 1) * 16 + col];
            y = __builtin_amdgcn_wmma_f32_16x16x4_f32(
                    false, a[c], false, bt, (short)0, y, false, false);
        }
        const int src = k + (lane & 16);  // broadcast w[row][k] within each half
        #pragma unroll
        for (int v = 0; v < 8; ++v) {
            float wk = __shfl(w[v], src, 32);
            acc[v] = fmaf(wk, y[v], acc[v]);
        }
    }

    // ---- store tile ----
    #pragma unroll
    for (int v = 0; v < 8; ++v) {
        ob[(size_t)(v + 8 * half) * 16 + col] = acc[v];
    }
}

// ---------------------------------------------------------------------------
// Launch
// ---------------------------------------------------------------------------
extern "C" void kernel_launch(void* const* d_in, const int* in_sizes, int n_in,
                              void* d_out, int out_size, void* d_ws, size_t ws_size,
                              hipStream_t stream) {
    const float* cpu_state   = (const float*)d_in[0];
    const float* ctrl_cursor = (const float*)d_in[1];
    const float* rule_memory = (const float*)d_in[2];
    const float* templates   = (const float*)d_in[3];
    const float* w1          = (const float*)d_in[4];
    const float* b1          = (const float*)d_in[5];
    const float* w2          = (const float*)d_in[6];
    const float* b2          = (const float*)d_in[7];
    const float* cell_w      = (const float*)d_in[8];
    const float* cell_b      = (const float*)d_in[9];
    const float* rsp_w       = (const float*)d_in[10];
    const float* rsp_b       = (const float*)d_in[11];
    const float* rp_w        = (const float*)d_in[12];
    const float* rp_b        = (const float*)d_in[13];
    float* outp = (float*)d_out;

    // workspace layout (floats)
    float* ws        = (float*)d_ws;
    float* cpu_sum   = ws;                       // B*16
    float* rule_sum  = cpu_sum  + GSS_B * 16;    // B*16
    float* colbias   = rule_sum + GSS_B * 16;    // B*16
    float* wpad      = colbias  + GSS_B * 16;    // 256
    float* Tmat      = wpad     + 256;           // B*8*256

    gss_reduce<<<GSS_B, 256, 0, stream>>>(cpu_state, rule_memory, cpu_sum, rule_sum);

    gss_prep<<<GSS_B, 256, 0, stream>>>(ctrl_cursor, templates, w1, b1, w2, b2,
                                        cell_w, cell_b, rsp_w, rsp_b, rp_w, rp_b,
                                        cpu_sum, rule_sum, Tmat, wpad, colbias);

    gss_main<<<GSS_B * BLKS_PER_B, WAVES_PER_BLK * 32, 0, stream>>>(
        cpu_state, Tmat, wpad, colbias, outp);
}